// PVMLayer_81346680586860
// MI455X (gfx1250) — compile-verified
//
#include <hip/hip_runtime.h>
#include <hip/hip_bf16.h>

// ---------------------------------------------------------------------------
// PVM layer (preLN -> in-proj -> 4x Mamba chunks -> postLN -> out-proj -> res)
// for MI455X / gfx1250.  All dense projections run on the matrix pipe via
// V_WMMA_F32_16X16X4_F32 (fp32-exact).  The selective scan is reduced to a
// barrier-free, reduction-free recurrence: dt-pre/B/C are precomputed by a
// fused conv+SiLU+GEMM (Wcomb = [w_dt . w_xp[:2] || w_xp[2:34]]), and the
// 64->32 output projection (+ residual) is hoisted into a post-scan GEMM.
// The scan is 1-step software-pipelined: step l+1's loads issue during step
// l's compute, so L0 latency is off the serial chain.
//
// Fast-path workspace (floats), needs ws_size >= ~321 MiB:
//   zc    : [64,4096, 32]  in-proj (chunk-permuted)      8388608 f
//   xz    : [64,4096,128]  u_pre || gate (pre-conv)     33554432 f
//   xd    : [64,4096, 96]  dtpre(64) | B(16) | C(16)    25165824 f
//   y     : [64,4096, 64]  scan output                  16777216 f
//   Wcomb : [96,64]                                         6144 f
//   z2    : [16,4096,128]  aliases xd (dead after scan)
// Fallback (ws_size >= 192 MiB): monolithic scan (LDS dot products).
// ---------------------------------------------------------------------------

typedef __attribute__((ext_vector_type(2))) float v2f;
typedef __attribute__((ext_vector_type(4))) float v4f;
typedef __attribute__((ext_vector_type(8))) float v8f;

#define BB   16
#define CC   256
#define LL   4096
#define BD   128
#define DM   32
#define DI   64
#define DSN  16
#define RR   2
#define NSEQ 64
#define NROWS (NSEQ * LL)   // 262144 mamba rows
#define XDW  96             // dtpre(64) + B(16) + C(16)

static __device__ __forceinline__ v8f wmma4(v2f a, v2f b, v8f c) {
  // D = A(16x4 f32) * B(4x16 f32) + C(16x16 f32)
  return __builtin_amdgcn_wmma_f32_16x16x4_f32(
      false, a, false, b, (short)0, c, false, false);
}

static __device__ __forceinline__ float siluf(float v) {
  return v / (1.f + __expf(-v));
}

// ---------------------------------------------------------------------------
// K0: build combined weight Wcomb[96][64].
//   n <  64 : sum_r w_dt[n][r] * w_xp[r][k]      (dt-pre, w_dt folded in)
//   64..79  : w_xp[2 + (n-64)][k]                (B rows)
//   80..95  : w_xp[18 + (n-80)][k]               (C rows)
// ---------------------------------------------------------------------------
__global__ __launch_bounds__(64) void k0_wcomb(
    const float* __restrict__ w_xp, const float* __restrict__ w_dt,
    float* __restrict__ wcomb) {
  const int n = blockIdx.x;   // 0..95
  const int k = threadIdx.x;  // 0..63
  float v;
  if (n < 64)       v = w_dt[n * 2 + 0] * w_xp[k] + w_dt[n * 2 + 1] * w_xp[DI + k];
  else if (n < 80)  v = w_xp[(2 + n - 64) * DI + k];
  else              v = w_xp[(18 + n - 80) * DI + k];
  wcomb[n * DI + k] = v;
}

// ---------------------------------------------------------------------------
// K1: pre-LN over C=256 + in-projection (256 -> 128) via WMMA.
// ---------------------------------------------------------------------------
#define LP1 257
__global__ __launch_bounds__(256) void k1_preln_inproj(
    const float* __restrict__ x, const float* __restrict__ g,
    const float* __restrict__ bta, const float* __restrict__ w_in,
    const float* __restrict__ b_in, float* __restrict__ zc) {
  __shared__ float As[16 * LP1];
  const int t  = threadIdx.x;
  const int p0 = blockIdx.x * 16;
  const int b  = p0 >> 12;
  const int l0 = p0 & (LL - 1);
  const float* xb = x + (size_t)b * CC * LL;

  for (int it = 0; it < 16; ++it) {
    int idx = t + it * 256;
    int c = idx >> 4, i = idx & 15;
    As[i * LP1 + c] = xb[(size_t)c * LL + l0 + i];
  }
  __syncthreads();

  const int r = t >> 4, q = t & 15;
  float s1 = 0.f, s2 = 0.f;
  for (int j = 0; j < 16; ++j) {
    float v = As[r * LP1 + q * 16 + j];
    s1 += v; s2 += v * v;
  }
  for (int m = 1; m < 16; m <<= 1) {
    s1 += __shfl_xor(s1, m, 32);
    s2 += __shfl_xor(s2, m, 32);
  }
  float mu   = s1 * (1.f / 256.f);
  float var  = s2 * (1.f / 256.f) - mu * mu;
  float rstd = rsqrtf(var + 1e-5f);
  for (int j = 0; j < 16; ++j) {
    int c = q * 16 + j;
    float v = As[r * LP1 + c];
    As[r * LP1 + c] = (v - mu) * rstd * g[c] + bta[c];
  }
  __syncthreads();

  const int lane = t & 31, wave = t >> 5;
  const int cr = lane & 15, hi = lane >> 4;
  const int n0 = wave * 16;
  v8f acc = {};
  for (int k0 = 0; k0 < CC; k0 += 4) {
    int kk = k0 + 2 * hi;
    v2f a, bb;
    a.x = As[cr * LP1 + kk];
    a.y = As[cr * LP1 + kk + 1];
    const float* wp = w_in + (size_t)(n0 + cr) * CC + kk;
    bb.x = wp[0]; bb.y = wp[1];
    acc = wmma4(a, bb, acc);
  }
  const int jcol  = n0 + cr;
  const float bia = b_in[jcol];
  const int chunk = jcol >> 5, dm = jcol & 31;
  for (int rr = 0; rr < 8; ++rr) {
    int M = rr + 8 * hi;
    int l = (p0 + M) & (LL - 1);
    zc[(((size_t)(chunk * BB + b)) * LL + l) * DM + dm] = acc[rr] + bia;
  }
}

// ---------------------------------------------------------------------------
// K2: xz projection (32 -> 128) via WMMA.
// ---------------------------------------------------------------------------
__global__ __launch_bounds__(256) void k2_xzproj(
    const float* __restrict__ zc, const float* __restrict__ w_xz,
    float* __restrict__ xz) {
  const int t = threadIdx.x;
  const size_t row0 = (size_t)blockIdx.x * 16;
  const int lane = t & 31, wave = t >> 5;
  const int cr = lane & 15, hi = lane >> 4;
  const int n0 = wave * 16;
  const float* arow = zc + (row0 + cr) * DM;
  v8f acc = {};
  for (int k0 = 0; k0 < DM; k0 += 4) {
    int kk = k0 + 2 * hi;
    v2f a, bb;
    a.x = arow[kk]; a.y = arow[kk + 1];
    const float* wp = w_xz + (size_t)(n0 + cr) * DM + kk;
    bb.x = wp[0]; bb.y = wp[1];
    acc = wmma4(a, bb, acc);
  }
  for (int rr = 0; rr < 8; ++rr) {
    size_t M = row0 + rr + 8 * hi;
    xz[M * (2 * DI) + n0 + cr] = acc[rr];
  }
}

// ---------------------------------------------------------------------------
// K3: fused depthwise-conv + SiLU + GEMM (u @ Wcomb^T, K=64, N=96).
// 16-row tile per workgroup; A-tile (u) built from a 19-row u_pre halo.
// ---------------------------------------------------------------------------
#define UP  64   // upre_s row stride (dcol-fast, conflict-free)
#define USP 65   // u_s row stride    (row-fast WMMA reads, conflict-free)
__global__ __launch_bounds__(256) void k3_convxd(
    const float* __restrict__ xz, const float* __restrict__ wcomb,
    const float* __restrict__ conv_w, const float* __restrict__ conv_b,
    float* __restrict__ xd) {
  __shared__ float upre_s[19 * UP];
  __shared__ float u_s[16 * USP];
  __shared__ float cws[DI * 4];
  __shared__ float cbs[DI];
  const int t = threadIdx.x;
  const size_t row0 = (size_t)blockIdx.x * 16;
  const int n  = (int)(row0 >> 12);
  const int l0 = (int)(row0 & (LL - 1));
  const float* xzn = xz + (size_t)n * LL * (2 * DI);

  cws[t] = conv_w[t & 255];
  if (t < DI) cbs[t] = conv_b[t];
  for (int idx = t; idx < 19 * DI; idx += 256) {
    int jr = idx >> 6, dcol = idx & 63;
    int l = l0 - 3 + jr;
    upre_s[jr * UP + dcol] = (l >= 0) ? xzn[(size_t)l * (2 * DI) + dcol] : 0.f;
  }
  __syncthreads();

  for (int idx = t; idx < 16 * DI; idx += 256) {
    int i = idx >> 6, dcol = idx & 63;
    const float* up = upre_s + i * UP + dcol;
    float uc = cws[dcol * 4 + 0] * up[0 * UP] + cws[dcol * 4 + 1] * up[1 * UP] +
               cws[dcol * 4 + 2] * up[2 * UP] + cws[dcol * 4 + 3] * up[3 * UP] +
               cbs[dcol];
    u_s[i * USP + dcol] = siluf(uc);
  }
  __syncthreads();

  const int lane = t & 31, wave = t >> 5;
  if (wave < 6) {
    const int cr = lane & 15, hi = lane >> 4;
    const int n0 = wave * 16;
    v8f acc = {};
    for (int k0 = 0; k0 < DI; k0 += 4) {
      int kk = k0 + 2 * hi;
      v2f a, bb;
      a.x = u_s[cr * USP + kk];
      a.y = u_s[cr * USP + kk + 1];
      const float* wp = wcomb + (size_t)(n0 + cr) * DI + kk;
      bb.x = wp[0]; bb.y = wp[1];
      acc = wmma4(a, bb, acc);
    }
    for (int rr = 0; rr < 8; ++rr) {
      size_t row = row0 + rr + 8 * hi;
      xd[row * XDW + n0 + cr] = acc[rr];
    }
  }
}

// ---------------------------------------------------------------------------
// K4: lightweight scan — no LDS, no barriers, no cross-channel reductions,
// 1-step software pipeline (loads for l+1 issue during compute of l).
// One workgroup per sequence; thread d owns channel d with h[16] in regs.
// ---------------------------------------------------------------------------
__global__ __launch_bounds__(64) void k4_scan_lite(
    const float* __restrict__ xz, const float* __restrict__ xd,
    const float* __restrict__ conv_w, const float* __restrict__ conv_b,
    const float* __restrict__ b_dt, const float* __restrict__ A_log,
    const float* __restrict__ Dskip, float* __restrict__ ybuf) {
  const int d = threadIdx.x;
  const int n = blockIdx.x;
  const float cw0 = conv_w[d * 4 + 0], cw1 = conv_w[d * 4 + 1];
  const float cw2 = conv_w[d * 4 + 2], cw3 = conv_w[d * 4 + 3];
  const float cb  = conv_b[d];
  const float bdt = b_dt[d];
  const float dsk = Dskip[d];
  float areg[DSN];
  for (int s = 0; s < DSN; ++s) areg[s] = -__expf(A_log[d * DSN + s]);
  float h[DSN];
  for (int s = 0; s < DSN; ++s) h[s] = 0.f;
  float r0 = 0.f, r1 = 0.f, r2 = 0.f;

  const float* xzn = xz + (size_t)n * LL * (2 * DI);
  const float* xdn = xd + (size_t)n * LL * XDW;
  float* yn = ybuf + (size_t)n * LL * DI;

  // pipeline prologue: load step 0
  float up  = xzn[d];
  float gg  = xzn[DI + d];
  float dtp = xdn[d];
  const v4f* bc0 = (const v4f*)(xdn + 64);   // 16B-aligned: stride 96f, +64f
  v4f B0 = bc0[0], B1 = bc0[1], B2 = bc0[2], B3 = bc0[3];
  v4f C0 = bc0[4], C1 = bc0[5], C2 = bc0[6], C3 = bc0[7];

  for (int l = 0; l < LL; ++l) {
    // issue next step's loads first (independent of this step's compute)
    const int ln = (l + 1 < LL) ? (l + 1) : l;
    const float* nrow  = xzn + (size_t)ln * (2 * DI);
    const float* nrowx = xdn + (size_t)ln * XDW;
    float up_n  = nrow[d];
    float gg_n  = nrow[DI + d];
    float dtp_n = nrowx[d];
    const v4f* nbc = (const v4f*)(nrowx + 64);
    v4f B0n = nbc[0], B1n = nbc[1], B2n = nbc[2], B3n = nbc[3];
    v4f C0n = nbc[4], C1n = nbc[5], C2n = nbc[6], C3n = nbc[7];
    __builtin_prefetch(nrow + 8 * (2 * DI), 0, 1);
    __builtin_prefetch(nrowx + 8 * XDW, 0, 1);

    // compute step l with previously loaded registers
    float uc = cw0 * r0 + cw1 * r1 + cw2 * r2 + cw3 * up + cb;
    r0 = r1; r1 = r2; r2 = up;
    float u  = siluf(uc);
    float gs = siluf(gg);
    float sp = dtp + bdt;
    float dt = (sp > 15.f) ? sp : __logf(1.f + __expf(sp));
    float du = dt * u;

    float y0 = 0.f, y1 = 0.f, y2 = 0.f, y3 = 0.f;
    #pragma unroll
    for (int s = 0; s < 4; ++s) {
      h[s]      = __expf(dt * areg[s])      * h[s]      + du * B0[s];
      h[s + 4]  = __expf(dt * areg[s + 4])  * h[s + 4]  + du * B1[s];
      h[s + 8]  = __expf(dt * areg[s + 8])  * h[s + 8]  + du * B2[s];
      h[s + 12] = __expf(dt * areg[s + 12]) * h[s + 12] + du * B3[s];
      y0 += h[s] * C0[s];
      y1 += h[s + 4] * C1[s];
      y2 += h[s + 8] * C2[s];
      y3 += h[s + 12] * C3[s];
    }
    float y = ((y0 + y1) + (y2 + y3) + u * dsk) * gs;
    yn[(size_t)l * DI + d] = y;

    // rotate pipeline registers
    up = up_n; gg = gg_n; dtp = dtp_n;
    B0 = B0n; B1 = B1n; B2 = B2n; B3 = B3n;
    C0 = C0n; C1 = C1n; C2 = C2n; C3 = C3n;
  }
}

// ---------------------------------------------------------------------------
// K5: output projection (64 -> 32) via WMMA + zc residual, chunk-permuted
// store into z2[B, L, 128].
// ---------------------------------------------------------------------------
__global__ __launch_bounds__(64) void k5_outproj(
    const float* __restrict__ ybuf, const float* __restrict__ w_out,
    const float* __restrict__ zc, float* __restrict__ z2) {
  const int t = threadIdx.x;
  const size_t row0 = (size_t)blockIdx.x * 16;
  const int n = (int)(row0 >> 12);
  const int chunk = n >> 4, bix = n & 15;
  const int lane = t & 31, wave = t >> 5;
  const int cr = lane & 15, hi = lane >> 4;
  const int n0 = wave * 16;
  const float* arow = ybuf + (row0 + cr) * DI;
  v8f acc = {};
  for (int k0 = 0; k0 < DI; k0 += 4) {
    int kk = k0 + 2 * hi;
    v2f a, bb;
    a.x = arow[kk]; a.y = arow[kk + 1];
    const float* wp = w_out + (size_t)(n0 + cr) * DI + kk;
    bb.x = wp[0]; bb.y = wp[1];
    acc = wmma4(a, bb, acc);
  }
  const int j = n0 + cr;
  for (int rr = 0; rr < 8; ++rr) {
    size_t row = row0 + rr + 8 * hi;
    int l = (int)(row & (LL - 1));
    float m = acc[rr] + zc[row * DM + j];
    z2[((size_t)bix * LL + l) * BD + chunk * DM + j] = m;
  }
}

// ---------------------------------------------------------------------------
// K3m (fallback, small ws): monolithic scan with LDS dot products.
// ---------------------------------------------------------------------------
__global__ __launch_bounds__(64) void k3_scan_mono(
    const float* __restrict__ xz, const float* __restrict__ zc,
    const float* __restrict__ conv_w, const float* __restrict__ conv_b,
    const float* __restrict__ w_xp, const float* __restrict__ w_dt,
    const float* __restrict__ b_dt, const float* __restrict__ A_log,
    const float* __restrict__ Dskip, const float* __restrict__ w_out,
    float* __restrict__ z2) {
  __shared__ float wxp_s[34 * 65];
  __shared__ float wout_s[DM * 65];
  __shared__ float u_sh[DI];
  __shared__ float xd_sh[34];
  __shared__ float y_sh[DI];
  const int d = threadIdx.x;
  const int n = blockIdx.x;
  for (int i = d; i < 34 * DI; i += 64) wxp_s[(i >> 6) * 65 + (i & 63)] = w_xp[i];
  for (int i = d; i < DM * DI; i += 64) wout_s[(i >> 6) * 65 + (i & 63)] = w_out[i];
  const float cw0 = conv_w[d * 4 + 0], cw1 = conv_w[d * 4 + 1];
  const float cw2 = conv_w[d * 4 + 2], cw3 = conv_w[d * 4 + 3];
  const float cb  = conv_b[d];
  const float wd0 = w_dt[d * 2 + 0], wd1 = w_dt[d * 2 + 1], bdt = b_dt[d];
  const float dsk = Dskip[d];
  float areg[DSN];
  for (int s = 0; s < DSN; ++s) areg[s] = -__expf(A_log[d * DSN + s]);
  float h[DSN];
  for (int s = 0; s < DSN; ++s) h[s] = 0.f;
  float r0 = 0.f, r1 = 0.f, r2 = 0.f;
  const float* xzn = xz + (size_t)n * LL * (2 * DI);
  const float* zcn = zc + (size_t)n * LL * DM;
  const int chunk = n >> 4, bix = n & 15;
  float* z2n = z2 + (size_t)bix * LL * BD + chunk * DM;
  __syncthreads();
  for (int l = 0; l < LL; ++l) {
    const float* row = xzn + (size_t)l * (2 * DI);
    float up = row[d], gg = row[DI + d];
    __builtin_prefetch(row + 8 * (2 * DI), 0, 1);
    float uc = cw0 * r0 + cw1 * r1 + cw2 * r2 + cw3 * up + cb;
    r0 = r1; r1 = r2; r2 = up;
    float u  = siluf(uc);
    float gs = siluf(gg);
    u_sh[d] = u;
    __syncthreads();
    if (d < 34) {
      float acc = 0.f;
      const float* wr = wxp_s + d * 65;
      #pragma unroll 8
      for (int j = 0; j < DI; ++j) acc += u_sh[j] * wr[j];
      xd_sh[d] = acc;
    }
    __syncthreads();
    float sp = xd_sh[0] * wd0 + xd_sh[1] * wd1 + bdt;
    float dt = (sp > 15.f) ? sp : __logf(1.f + __expf(sp));
    float du = dt * u;
    float y = 0.f;
    #pragma unroll
    for (int s = 0; s < DSN; ++s) {
      h[s] = __expf(dt * areg[s]) * h[s] + du * xd_sh[RR + s];
      y += h[s] * xd_sh[RR + DSN + s];
    }
    y = (y + u * dsk) * gs;
    y_sh[d] = y;
    __syncthreads();
    if (d < DM) {
      float acc = 0.f;
      const float* wr = wout_s + d * 65;
      #pragma unroll 8
      for (int j = 0; j < DI; ++j) acc += y_sh[j] * wr[j];
      z2n[(size_t)l * BD + d] = acc + zcn[(size_t)l * DM + d];
    }
  }
}

// ---------------------------------------------------------------------------
// K6: post-LN over 128 + bo projection (128 -> 256) via WMMA + residual.
// ---------------------------------------------------------------------------
#define LP4 129
__global__ __launch_bounds__(256) void k6_postln_bo(
    const float* __restrict__ z2, const float* __restrict__ g,
    const float* __restrict__ bta, const float* __restrict__ w_bo,
    const float* __restrict__ b_bo, const float* __restrict__ rs_ptr,
    const float* __restrict__ x, float* __restrict__ out) {
  __shared__ float As[16 * LP4];
  const int t  = threadIdx.x;
  const int p0 = blockIdx.x * 16;
  const int b  = p0 >> 12;
  const int l0 = p0 & (LL - 1);
  for (int it = 0; it < 8; ++it) {
    int idx = t + it * 256;
    int rr = idx >> 7, c = idx & 127;
    As[rr * LP4 + c] = z2[((size_t)(p0 + rr)) * BD + c];
  }
  __syncthreads();
  const int r = t >> 4, q = t & 15;
  float s1 = 0.f, s2 = 0.f;
  for (int j = 0; j < 8; ++j) {
    float v = As[r * LP4 + q * 8 + j];
    s1 += v; s2 += v * v;
  }
  for (int m = 1; m < 16; m <<= 1) {
    s1 += __shfl_xor(s1, m, 32);
    s2 += __shfl_xor(s2, m, 32);
  }
  float mu   = s1 * (1.f / 128.f);
  float var  = s2 * (1.f / 128.f) - mu * mu;
  float rstd = rsqrtf(var + 1e-5f);
  for (int j = 0; j < 8; ++j) {
    int c = q * 8 + j;
    float v = As[r * LP4 + c];
    As[r * LP4 + c] = (v - mu) * rstd * g[c] + bta[c];
  }
  __syncthreads();
  const float rsc = *rs_ptr;
  const int lane = t & 31, wave = t >> 5;
  const int cr = lane & 15, hi = lane >> 4;
  for (int half = 0; half < 2; ++half) {
    const int n0 = wave * 16 + half * 128;
    v8f acc = {};
    for (int k0 = 0; k0 < BD; k0 += 4) {
      int kk = k0 + 2 * hi;
      v2f a, bb;
      a.x = As[cr * LP4 + kk];
      a.y = As[cr * LP4 + kk + 1];
      const float* wp = w_bo + (size_t)(n0 + cr) * BD + kk;
      bb.x = wp[0]; bb.y = wp[1];
      acc = wmma4(a, bb, acc);
    }
    const int cch = n0 + cr;
    const float bia = b_bo[cch];
    for (int rr2 = 0; rr2 < 8; ++rr2) {
      int M = rr2 + 8 * hi;
      int l = l0 + M;
      size_t oidx = ((size_t)b * CC + cch) * LL + l;
      out[oidx] = x[oidx] + (acc[rr2] + bia) * rsc;
    }
  }
}

// ---------------------------------------------------------------------------
extern "C" void kernel_launch(void* const* d_in, const int* in_sizes, int n_in,
                              void* d_out, int out_size, void* d_ws, size_t ws_size,
                              hipStream_t stream) {
  const float* x       = (const float*)d_in[0];
  const float* preln_g = (const float*)d_in[1];
  const float* preln_b = (const float*)d_in[2];
  const float* w_in    = (const float*)d_in[3];
  const float* b_in    = (const float*)d_in[4];
  const float* w_xz    = (const float*)d_in[5];
  const float* conv_w  = (const float*)d_in[6];
  const float* conv_b  = (const float*)d_in[7];
  const float* w_xp    = (const float*)d_in[8];
  const float* w_dt    = (const float*)d_in[9];
  const float* b_dt    = (const float*)d_in[10];
  const float* A_log   = (const float*)d_in[11];
  const float* Dskip   = (const float*)d_in[12];
  const float* w_out   = (const float*)d_in[13];
  const float* postln_g= (const float*)d_in[14];
  const float* postln_b= (const float*)d_in[15];
  const float* w_bo    = (const float*)d_in[16];
  const float* b_bo    = (const float*)d_in[17];
  const float* rscale  = (const float*)d_in[18];

  float* ws = (float*)d_ws;
  const size_t ZC = (size_t)NSEQ * LL * DM;        //  8388608
  const size_t XZ = (size_t)NSEQ * LL * 2 * DI;    // 33554432
  const size_t XD = (size_t)NROWS * XDW;           // 25165824
  const size_t YB = (size_t)NROWS * DI;            // 16777216
  const size_t WC = 96 * DI;                       //     6144

  float* zc = ws;
  float* xz = zc + ZC;
  const size_t need_fast = (ZC + XZ + XD + YB + WC) * sizeof(float);  // ~321 MiB
  const bool fast = (ws_size >= need_fast);

  k1_preln_inproj<<<(BB * LL) / 16, 256, 0, stream>>>(x, preln_g, preln_b,
                                                      w_in, b_in, zc);
  k2_xzproj<<<NROWS / 16, 256, 0, stream>>>(zc, w_xz, xz);

  if (fast) {
    float* xd    = xz + XZ;
    float* ybuf  = xd + XD;
    float* wcomb = ybuf + YB;
    float* z2    = xd;  // alias: xd dead once scan finishes
    k0_wcomb<<<96, 64, 0, stream>>>(w_xp, w_dt, wcomb);
    k3_convxd<<<NROWS / 16, 256, 0, stream>>>(xz, wcomb, conv_w, conv_b, xd);
    k4_scan_lite<<<NSEQ, 64, 0, stream>>>(xz, xd, conv_w, conv_b, b_dt, A_log,
                                          Dskip, ybuf);
    k5_outproj<<<NROWS / 16, 64, 0, stream>>>(ybuf, w_out, zc, z2);
    k6_postln_bo<<<(BB * LL) / 16, 256, 0, stream>>>(z2, postln_g, postln_b,
                                                     w_bo, b_bo, rscale, x,
                                                     (float*)d_out);
  } else {
    float* z2 = xz + XZ;  // compact layout (~192 MiB total)
    k3_scan_mono<<<NSEQ, 64, 0, stream>>>(xz, zc, conv_w, conv_b, w_xp, w_dt,
                                          b_dt, A_log, Dskip, w_out, z2);
    k6_postln_bo<<<(BB * LL) / 16, 256, 0, stream>>>(z2, postln_g, postln_b,
                                                     w_bo, b_bo, rscale, x,
                                                     (float*)d_out);
  }
}